// GetCovMatFromChol_54941221650671
// MI455X (gfx1250) — compile-verified
//
#include <hip/hip_runtime.h>
#include <stdint.h>

// GetCovMatFromChol: Q = L @ L^T for batched packed lower-triangular 3x3 L.
// Pure streaming kernel: 24B in / 36B out per element, ~0.23 FLOP/byte.
// Strategy: async global->LDS staged, fully coalesced b128 loads; b128 NT stores.

typedef float v4f __attribute__((ext_vector_type(4)));   // native vector: OK for nontemporal builtins

#define BLOCK 256
#define EPT 4                        // elements per thread
#define TILE_ELEMS (BLOCK * EPT)     // 1024 elements per block
#define TILE_FLOATS (TILE_ELEMS * 6) // 6144 floats = 24 KB LDS

__global__ __launch_bounds__(BLOCK)
void cov_from_chol_kernel(const float* __restrict__ chol,
                          float* __restrict__ out,
                          int n_elems) {
    __shared__ __align__(16) float tile[TILE_FLOATS];

    const int tid = threadIdx.x;
    const long long base = (long long)blockIdx.x * TILE_ELEMS; // first element of this block
    const int remaining = (int)((long long)n_elems - base);    // uniform per block

    if (remaining >= TILE_ELEMS) {
        // ---- Fast path: full tile ----
        const float* gsrc = chol + base * 6;                   // 24 KB, 16B aligned
        const uint32_t lds_base = (uint32_t)(uintptr_t)(&tile[0]);

        // 6 x b128 per thread: lane-consecutive 16B chunks -> perfectly coalesced.
        // gfx1250 async copy path (ASYNCcnt), GVS addressing: saddr(SGPR64) + vaddr(VGPR32).
        #pragma unroll
        for (int i = 0; i < 6; ++i) {
            uint32_t goff  = (uint32_t)((i * BLOCK + tid) * 16); // byte offset into tile
            uint32_t laddr = lds_base + goff;                    // LDS byte address
            asm volatile(
                "global_load_async_to_lds_b128 %0, %1, %2 offset:0"
                :
                : "v"(laddr), "v"(goff), "s"(gsrc)
                : "memory");
        }
        asm volatile("s_wait_asynccnt 0" ::: "memory");
        __syncthreads();

        // Each thread consumes 4 contiguous elements = 24 floats = 6 x b128 from LDS.
        float r[24];
        const v4f* lp = reinterpret_cast<const v4f*>(&tile[tid * 24]);
        #pragma unroll
        for (int i = 0; i < 6; ++i) {
            v4f v = lp[i];                       // ds_load_b128
            r[4 * i + 0] = v.x;
            r[4 * i + 1] = v.y;
            r[4 * i + 2] = v.z;
            r[4 * i + 3] = v.w;
        }

        // Compute Q = L L^T per element.
        float o[36];
        #pragma unroll
        for (int k = 0; k < EPT; ++k) {
            const float a = r[6 * k + 0];
            const float b = r[6 * k + 1];
            const float c = r[6 * k + 2];
            const float d = r[6 * k + 3];
            const float e = r[6 * k + 4];
            const float f = r[6 * k + 5];
            const float q01 = a * b;
            const float q02 = a * d;
            const float q12 = b * d + c * e;
            o[9 * k + 0] = a * a;
            o[9 * k + 1] = q01;
            o[9 * k + 2] = q02;
            o[9 * k + 3] = q01;
            o[9 * k + 4] = b * b + c * c;
            o[9 * k + 5] = q12;
            o[9 * k + 6] = q02;
            o[9 * k + 7] = q12;
            o[9 * k + 8] = d * d + e * e + f * f;
        }

        // 36 floats per thread, 144B-aligned contiguous chunk -> 9 x b128 NT stores.
        v4f* op = reinterpret_cast<v4f*>(out + (base + (long long)tid * EPT) * 9);
        #pragma unroll
        for (int j = 0; j < 9; ++j) {
            v4f v = { o[4 * j + 0], o[4 * j + 1], o[4 * j + 2], o[4 * j + 3] };
            __builtin_nontemporal_store(v, &op[j]);  // 288 MB output stream > 192 MB L2
        }
    } else if (remaining > 0) {
        // ---- Tail path (never taken for n = 2^23, kept for generality) ----
        for (int e = tid; e < remaining; e += BLOCK) {
            const float* s = chol + (base + e) * 6;
            const float a = s[0], b = s[1], c = s[2], d = s[3], ee = s[4], f = s[5];
            float* q = out + (base + e) * 9;
            const float q01 = a * b;
            const float q02 = a * d;
            const float q12 = b * d + c * ee;
            q[0] = a * a;
            q[1] = q01;
            q[2] = q02;
            q[3] = q01;
            q[4] = b * b + c * c;
            q[5] = q12;
            q[6] = q02;
            q[7] = q12;
            q[8] = d * d + ee * ee + f * f;
        }
    }
}

extern "C" void kernel_launch(void* const* d_in, const int* in_sizes, int n_in,
                              void* d_out, int out_size, void* d_ws, size_t ws_size,
                              hipStream_t stream) {
    (void)n_in; (void)out_size; (void)d_ws; (void)ws_size;
    const float* chol = (const float*)d_in[0];
    float* out = (float*)d_out;
    const int n_elems = in_sizes[0] / 6;                       // 8388608
    const int grid = (n_elems + TILE_ELEMS - 1) / TILE_ELEMS;  // 8192 blocks
    cov_from_chol_kernel<<<grid, BLOCK, 0, stream>>>(chol, out, n_elems);
}